// ARNN_B_46952582480191
// MI455X (gfx1250) — compile-verified
//
#include <hip/hip_runtime.h>

#define HDIM 4096
#define NBLK 256        // HDIM/16 row-tiles
#define TPB  256        // 8 waves of 32
#define MAX_STEPS 16
#define EPS_F 0.01f

typedef __attribute__((ext_vector_type(2))) float v2f;
typedef __attribute__((ext_vector_type(8))) float v8f;

struct Ctrl { float cum; int done; int N; int pad; };

// D = A(16x4) * B(4x16) + C, f32. A: lanes 0-15 hold rows M, VGPR0/1 = K0/K1,
// lanes 16-31 = K2/K3. B columns are made identical (h broadcast), so every
// column of D carries the 16 row-partials; rows 0-7 live in lane 0's 8 acc
// VGPRs, rows 8-15 in lane 16's.
__device__ __forceinline__ v8f wmma_f32_4(v2f a, v2f b, v8f c) {
  return __builtin_amdgcn_wmma_f32_16x16x4_f32(
      /*neg_a=*/false, a, /*neg_b=*/false, b,
      /*c_mod=*/(short)0, c, /*reuse_a=*/false, /*reuse_b=*/false);
}

__global__ __launch_bounds__(TPB) void init_kernel(const float* __restrict__ s,
                                                   float* __restrict__ h0,
                                                   float* __restrict__ partials,
                                                   Ctrl* __restrict__ ctrl) {
  int idx = blockIdx.x * TPB + threadIdx.x;
  if (idx < HDIM) h0[idx] = s[idx];
  if (idx < NBLK) partials[idx] = 0.f;
  if (idx == 0) { ctrl->cum = 0.f; ctrl->done = 0; ctrl->N = MAX_STEPS - 1; }
}

// z_x[row] = b_ih + b_hh + W_ih[row, 1:] @ x   (stride HDIM+1, skip flag col)
__global__ __launch_bounds__(TPB) void zx_kernel(const float* __restrict__ Wih,
                                                 const float* __restrict__ x,
                                                 const float* __restrict__ b_ih,
                                                 const float* __restrict__ b_hh,
                                                 float* __restrict__ zx) {
  __shared__ float sh_x[HDIM];
  __shared__ float sh_rows[8][16];
  for (int i = threadIdx.x; i < HDIM; i += TPB) sh_x[i] = x[i];
  __syncthreads();

  const int lane = threadIdx.x & 31, wave = threadIdx.x >> 5;
  const int m = lane & 15, khalf = lane >> 4;
  const float* wrow = Wih + (size_t)(blockIdx.x * 16 + m) * (HDIM + 1) + 1;
  const int k0 = wave * (HDIM / 8);
  v8f acc = {};
  for (int k = k0; k < k0 + HDIM / 8; k += 4) {
    const int kk = k + khalf * 2;
    v2f a; a.x = wrow[kk]; a.y = wrow[kk + 1];
    v2f b; b.x = sh_x[kk]; b.y = sh_x[kk + 1];
    acc = wmma_f32_4(a, b, acc);
  }
  if (lane == 0 || lane == 16) {
    float* dst = &sh_rows[wave][khalf * 8];
#pragma unroll
    for (int i = 0; i < 8; ++i) dst[i] = acc[i];
  }
  __syncthreads();
  if (threadIdx.x < 16) {
    const int r = threadIdx.x, row = blockIdx.x * 16 + r;
    float sum = 0.f;
#pragma unroll
    for (int w = 0; w < 8; ++w) sum += sh_rows[w][r];
    zx[row] = sum + b_ih[row] + b_hh[row];
  }
}

// h_out = tanh(z_x + (t==0 ? W_ih[:,0] : 0) + W_hh @ h_in); also per-block
// partial of w_halt . h_out. Early-outs once the halting loop has crossed.
__global__ __launch_bounds__(TPB) void rnn_step_kernel(
    const float* __restrict__ Whh, const float* __restrict__ Wih,
    const float* __restrict__ zx, const float* __restrict__ w_halt,
    const float* __restrict__ h_in, float* __restrict__ h_out,
    float* __restrict__ partials, const Ctrl* __restrict__ ctrl, int t) {
  if (ctrl->done) return;
  __shared__ float sh_h[HDIM];
  __shared__ float sh_rows[8][16];
  for (int i = threadIdx.x; i < HDIM; i += TPB) sh_h[i] = h_in[i];
  __syncthreads();

  const int lane = threadIdx.x & 31, wave = threadIdx.x >> 5;
  const int m = lane & 15, khalf = lane >> 4;
  const float* wrow = Whh + (size_t)(blockIdx.x * 16 + m) * HDIM;
  const int k0 = wave * (HDIM / 8);
  v8f acc = {};
  for (int k = k0; k < k0 + HDIM / 8; k += 4) {
    const int kk = k + khalf * 2;
    v2f a; a.x = wrow[kk]; a.y = wrow[kk + 1];
    v2f b; b.x = sh_h[kk]; b.y = sh_h[kk + 1];
    acc = wmma_f32_4(a, b, acc);
  }
  if (lane == 0 || lane == 16) {
    float* dst = &sh_rows[wave][khalf * 8];
#pragma unroll
    for (int i = 0; i < 8; ++i) dst[i] = acc[i];
  }
  __syncthreads();
  if (threadIdx.x < 16) {
    const int r = threadIdx.x, row = blockIdx.x * 16 + r;
    float sum = 0.f;
#pragma unroll
    for (int w = 0; w < 8; ++w) sum += sh_rows[w][r];
    float val = zx[row] + sum;
    if (t == 0) val += Wih[(size_t)row * (HDIM + 1)];  // flag column
    float hn = tanhf(val);
    h_out[row] = hn;
    sh_rows[0][r] = w_halt[row] * hn;
  }
  __syncthreads();
  if (threadIdx.x == 0) {
    float s = 0.f;
#pragma unroll
    for (int r = 0; r < 16; ++r) s += sh_rows[0][r];
    partials[blockIdx.x] = s;
  }
}

// Deterministic fixed-order reduction of block partials; halting update.
__global__ __launch_bounds__(TPB) void halt_kernel(
    const float* __restrict__ partials, const float* __restrict__ b_halt,
    const float* __restrict__ h_new, float* __restrict__ h_final,
    Ctrl* __restrict__ ctrl, int t) {
  if (ctrl->done) return;
  __shared__ float red[TPB];
  __shared__ int take;
  red[threadIdx.x] = partials[threadIdx.x];
  __syncthreads();
  for (int s = TPB / 2; s > 0; s >>= 1) {
    if (threadIdx.x < s) red[threadIdx.x] += red[threadIdx.x + s];
    __syncthreads();
  }
  if (threadIdx.x == 0) {
    float p = 1.f / (1.f + expf(-(red[0] + b_halt[0])));
    float cum = ctrl->cum + p;
    ctrl->cum = cum;
    int cross = (cum >= 1.0f - EPS_F) || (t == MAX_STEPS - 1);
    take = cross;
    if (cross) { ctrl->N = t; ctrl->done = 1; }
  }
  __syncthreads();
  if (take)
    for (int i = threadIdx.x; i < HDIM; i += TPB) h_final[i] = h_new[i];
}

// out = W_out @ h_final + b_out; also emit s_new = h_final and ponder = N.
__global__ __launch_bounds__(TPB) void out_kernel(
    const float* __restrict__ Wout, const float* __restrict__ b_out,
    const float* __restrict__ h_final, const Ctrl* __restrict__ ctrl,
    float* __restrict__ out) {
  __shared__ float sh_h[HDIM];
  __shared__ float sh_rows[8][16];
  for (int i = threadIdx.x; i < HDIM; i += TPB) sh_h[i] = h_final[i];
  __syncthreads();

  const int lane = threadIdx.x & 31, wave = threadIdx.x >> 5;
  const int m = lane & 15, khalf = lane >> 4;
  const float* wrow = Wout + (size_t)(blockIdx.x * 16 + m) * HDIM;
  const int k0 = wave * (HDIM / 8);
  v8f acc = {};
  for (int k = k0; k < k0 + HDIM / 8; k += 4) {
    const int kk = k + khalf * 2;
    v2f a; a.x = wrow[kk]; a.y = wrow[kk + 1];
    v2f b; b.x = sh_h[kk]; b.y = sh_h[kk + 1];
    acc = wmma_f32_4(a, b, acc);
  }
  if (lane == 0 || lane == 16) {
    float* dst = &sh_rows[wave][khalf * 8];
#pragma unroll
    for (int i = 0; i < 8; ++i) dst[i] = acc[i];
  }
  __syncthreads();
  if (threadIdx.x < 16) {
    const int r = threadIdx.x, row = blockIdx.x * 16 + r;
    float sum = 0.f;
#pragma unroll
    for (int w = 0; w < 8; ++w) sum += sh_rows[w][r];
    out[row] = sum + b_out[row];         // output
    out[HDIM + row] = sh_h[row];         // s_new
  }
  if (blockIdx.x == 0 && threadIdx.x == 0)
    out[2 * HDIM] = (float)ctrl->N;      // ponder
}

extern "C" void kernel_launch(void* const* d_in, const int* in_sizes, int n_in,
                              void* d_out, int out_size, void* d_ws, size_t ws_size,
                              hipStream_t stream) {
  const float* x      = (const float*)d_in[0];
  const float* s      = (const float*)d_in[1];
  const float* W_ih   = (const float*)d_in[2];   // (H, 1+I) row-major, stride 4097
  const float* b_ih   = (const float*)d_in[3];
  const float* W_hh   = (const float*)d_in[4];   // (H, H)
  const float* b_hh   = (const float*)d_in[5];
  const float* w_halt = (const float*)d_in[6];   // (1, H)
  const float* b_halt = (const float*)d_in[7];   // scalar
  const float* W_out  = (const float*)d_in[8];   // (O, H)
  const float* b_out  = (const float*)d_in[9];
  float* outp = (float*)d_out;

  float* ws       = (float*)d_ws;
  float* zx       = ws;                // 4096
  float* h0       = ws + 4096;         // 4096
  float* h1       = ws + 8192;         // 4096
  float* hf       = ws + 12288;        // 4096
  float* partials = ws + 16384;        // 256
  Ctrl*  ctrl     = (Ctrl*)(ws + 16640);

  init_kernel<<<(HDIM + TPB - 1) / TPB, TPB, 0, stream>>>(s, h0, partials, ctrl);
  zx_kernel<<<NBLK, TPB, 0, stream>>>(W_ih, x, b_ih, b_hh, zx);

  float* bufs[2] = { h0, h1 };
  for (int t = 0; t < MAX_STEPS; ++t) {
    float* hin  = bufs[t & 1];
    float* hout = bufs[(t + 1) & 1];
    rnn_step_kernel<<<NBLK, TPB, 0, stream>>>(W_hh, W_ih, zx, w_halt, hin, hout,
                                              partials, ctrl, t);
    halt_kernel<<<1, TPB, 0, stream>>>(partials, b_halt, hout, hf, ctrl, t);
  }
  out_kernel<<<NBLK, TPB, 0, stream>>>(W_out, b_out, hf, ctrl, outp);
}